// Policy_75076028334553
// MI455X (gfx1250) — compile-verified
//
#include <hip/hip_runtime.h>

// ---------------------------------------------------------------------------
// Policy net: linear1(16->4)+relu -> LSTM(H=4, T=256) -> linear2(4->8)+softmax
// B=4096 sequences. One wave32 owns 16 batch rows for the whole T loop.
//
// Per chunk of 8 time steps:
//   Stage  : x tile -> LDS via GLOBAL_LOAD_ASYNC_TO_LDS_B128 (ASYNCcnt path).
//            The LDS tile pointer is passed into the asm as an operand so the
//            address escapes: without that, GlobalOpt sees a never-stored LDS
//            global and folds the ds reads to undef ("m" constraints on LDS
//            crash the backend, so escape-via-operand is the supported route).
//   Phase A: 8 independent linear1 K-chains (4x V_WMMA_F32_16X16X4_F32 each)
//   Phase B: sequential LSTM recurrence, 2 WMMAs + activations per step
//   Phase C: 8 independent linear2 WMMAs + per-lane softmax + stores
// Lane exchanges are lane^16 ds_swizzle SWAPX16 executed at FULL EXEC, then
// selected with v_cndmask. Fast v_rcp_f32 for sigmoid/softmax.
// ---------------------------------------------------------------------------

typedef __attribute__((ext_vector_type(2))) float v2f;
typedef __attribute__((ext_vector_type(8))) float v8f;

#define WMMA4(A, Bv, Cv) \
  __builtin_amdgcn_wmma_f32_16x16x4_f32(false, (A), false, (Bv), (short)0, (Cv), false, false)

constexpr int kB  = 4096;  // batch
constexpr int kT  = 256;   // seq len
constexpr int kS  = 16;    // state dim (linear1 input)
constexpr int kA  = 8;     // action dim
constexpr int kH  = 4;     // hidden dim
constexpr int kTT = 8;     // time steps staged per LDS chunk
constexpr int kTile = 16 * kTT * kS;       // floats in the LDS x tile

__device__ __forceinline__ float swap16(float v) {   // value from lane^16
    return __int_as_float(__builtin_amdgcn_ds_swizzle(__float_as_int(v), 0x401f));
}
__device__ __forceinline__ float fast_rcp(float x) { // v_rcp_f32, ~1 ulp
    return __builtin_amdgcn_rcpf(x);
}
__device__ __forceinline__ float fast_sigmoid(float x) {
    return fast_rcp(1.f + __expf(-x));
}

__global__ __launch_bounds__(32)
void policy_lstm_kernel(const float* __restrict__ x,
                        const float* __restrict__ h0in,
                        const float* __restrict__ c0in,
                        const float* __restrict__ w1,  const float* __restrict__ b1,
                        const float* __restrict__ wih, const float* __restrict__ whh,
                        const float* __restrict__ bih, const float* __restrict__ bhh,
                        const float* __restrict__ w2,  const float* __restrict__ b2,
                        float* __restrict__ out)
{
    __shared__ float lds_x[kTile];           // [batch-in-tile][tt][s] = 8 KiB

    const int lane = threadIdx.x;            // 0..31 (wave32)
    const int nh   = lane & 15;              // column / batch-in-tile index
    const int hi   = lane >> 4;              // 0 = lanes 0-15, 1 = lanes 16-31
    const int b0   = blockIdx.x * 16;        // first batch row of this tile

    const void* ldsesc = (const void*)lds_x; // escape handle for the asm below

    // ---- A operands (16x4 f32: lane m holds cols {0,1} low-half / {2,3} high) ----
    v2f a1[4];                               // w1 padded to 16 rows, K-chained by 4
    #pragma unroll
    for (int k = 0; k < 4; ++k) {
        const int col = 4 * k + 2 * hi;
        a1[k].x = (nh < kH) ? w1[nh * kS + col]     : 0.f;
        a1[k].y = (nh < kH) ? w1[nh * kS + col + 1] : 0.f;
    }
    v2f aih, ahh, a2m;
    {
        const int col = 2 * hi;
        aih.x = wih[nh * kH + col];  aih.y = wih[nh * kH + col + 1];   // 16x4 exact
        ahh.x = whh[nh * kH + col];  ahh.y = whh[nh * kH + col + 1];   // 16x4 exact
        a2m.x = (nh < kA) ? w2[nh * kH + col]     : 0.f;               // 8x4 padded
        a2m.y = (nh < kA) ? w2[nh * kH + col + 1] : 0.f;
    }

    // ---- C bias operands (16x16 f32 D layout: VGPR r = row r | row r+8) ----
    v8f cb1, cbg, cb2;
    #pragma unroll
    for (int r = 0; r < 8; ++r) {
        cb1[r] = (!hi && r < kH) ? b1[r] : 0.f;
        cbg[r] = bih[r + 8 * hi] + bhh[r + 8 * hi];
        cb2[r] = hi ? 0.f : b2[r];
    }

    // ---- LSTM state: lanes 0-15 own batch nh; bh kept in WMMA-B layout ----
    float hs[4], cs[4];
    #pragma unroll
    for (int j = 0; j < 4; ++j) {
        hs[j] = h0in[(size_t)(b0 + nh) * kH + j];
        cs[j] = c0in[(size_t)(b0 + nh) * kH + j];
    }
    v2f bh;                                  // B layout: rows {0,1} low / {2,3} high
    bh.x = h0in[(size_t)(b0 + nh) * kH + 2 * hi];
    bh.y = h0in[(size_t)(b0 + nh) * kH + 2 * hi + 1];

    const size_t probs_elems = (size_t)kB * kT * kA;

    for (int tc = 0; tc < kT / kTT; ++tc) {
        const int t0 = tc * kTT;
        __syncthreads();                     // previous chunk fully consumed

        // ---- stage x[b0..b0+15][t0..t0+7][0..15] -> LDS, async-to-LDS path ----
        {
            const int row = lane >> 1, half = lane & 1;    // 2 lanes per batch row
            const float* gsrc = x + ((size_t)(b0 + row) * kT + t0) * kS + half * 64;
            const unsigned lds0 = (unsigned)((row * (kTT * kS) + half * 64) * sizeof(float));
            #pragma unroll
            for (int i = 0; i < 16; ++i) {                 // 16 x b128 = 256B/lane
                asm volatile("global_load_async_to_lds_b128 %0, %1, off offset:%2"
                             :
                             : "v"(lds0 + 16u * i), "v"(gsrc), "i"(16 * i), "s"(ldsesc)
                             : "memory");
            }
            if (tc + 1 < kT / kTT) {         // gfx1250 global_prefetch_b8 on next tile
                const char* np = (const char*)(x + ((size_t)(b0 + row) * kT + t0 + kTT) * kS) + half * 256;
                __builtin_prefetch(np,       0, 3);
                __builtin_prefetch(np + 128, 0, 3);
            }
            asm volatile("s_wait_asynccnt 0x0" :: "s"(ldsesc) : "memory");
        }
        __syncthreads();

        // ======== Phase A: linear1 for all 8 steps (independent WMMA chains) ====
        v2f bzb[kTT];
        #pragma unroll
        for (int tt = 0; tt < kTT; ++tt) {
            v8f acc = cb1;
            const float* xr = lds_x + nh * (kTT * kS) + tt * kS;
            #pragma unroll
            for (int k = 0; k < 4; ++k) {
                v2f bx;
                bx.x = xr[4 * k + 2 * hi];
                bx.y = xr[4 * k + 2 * hi + 1];
                acc = WMMA4(a1[k], bx, acc);
            }
            const float z0 = fmaxf(acc[0], 0.f), z1 = fmaxf(acc[1], 0.f);
            const float z2 = fmaxf(acc[2], 0.f), z3 = fmaxf(acc[3], 0.f);
            const float z2l = swap16(z2);            // full-EXEC cross-lane ops,
            const float z3l = swap16(z3);            // then uniform select
            v2f bz;
            bz.x = hi ? z2l : z0;
            bz.y = hi ? z3l : z1;
            bzb[tt] = bz;
        }

        // ======== Phase B: sequential LSTM recurrence (minimal critical path) ===
        v2f bhb[kTT];
        #pragma unroll
        for (int tt = 0; tt < kTT; ++tt) {
            v8f g = cbg;
            g = WMMA4(aih, bzb[tt], g);
            g = WMMA4(ahh, bh, g);

            // lanes 0-15: rows 0-7 = i,f (sigmoid). lanes 16-31: rows 8-15 = g,o
            // tanh(v) = 2*sigmoid(2v)-1 -> one exp + one rcp per value
            float act[8];
            #pragma unroll
            for (int r = 0; r < 8; ++r) {
                const float v   = g[r];
                const bool  tnh = (hi && r < 4);
                const float s   = fast_sigmoid(tnh ? 2.f * v : v);
                act[r] = tnh ? (2.f * s - 1.f) : s;
            }

            // c,h update on lanes 0-15 (pull tanh(g), sigmoid(o) down via swap16)
            #pragma unroll
            for (int j = 0; j < 4; ++j) {
                const float gj = swap16(act[j]);
                const float oj = swap16(act[j + 4]);
                cs[j] = act[j + 4] * cs[j] + act[j] * gj;              // f*c + i*g
                hs[j] = oj * (2.f * fast_sigmoid(2.f * cs[j]) - 1.f);  // o*tanh(c)
            }
            const float h2l = swap16(hs[2]);         // full-EXEC, then select
            const float h3l = swap16(hs[3]);
            bh.x = hi ? h2l : hs[0];
            bh.y = hi ? h3l : hs[1];
            bhb[tt] = bh;
        }

        // ======== Phase C: linear2 + softmax + store (independent per step) =====
        #pragma unroll
        for (int tt = 0; tt < kTT; ++tt) {
            v8f lo = WMMA4(a2m, bhb[tt], cb2);
            float m = lo[0];
            #pragma unroll
            for (int r = 1; r < 8; ++r) m = fmaxf(m, lo[r]);
            float e[8], sum = 0.f;
            #pragma unroll
            for (int r = 0; r < 8; ++r) { e[r] = __expf(lo[r] - m); sum += e[r]; }
            const float inv = fast_rcp(sum);
            if (!hi) {
                float* op = out + ((size_t)(b0 + nh) * kT + (t0 + tt)) * kA;
                float4 p0 = { e[0]*inv, e[1]*inv, e[2]*inv, e[3]*inv };
                float4 p1 = { e[4]*inv, e[5]*inv, e[6]*inv, e[7]*inv };
                ((float4*)op)[0] = p0;
                ((float4*)op)[1] = p1;
            }
        }
    }

    // ---- final hT, cT ----
    if (!hi) {
        float* hp = out + probs_elems + (size_t)(b0 + nh) * kH;
        float* cp = hp + (size_t)kB * kH;
        #pragma unroll
        for (int j = 0; j < 4; ++j) { hp[j] = hs[j]; cp[j] = cs[j]; }
    }
}

extern "C" void kernel_launch(void* const* d_in, const int* in_sizes, int n_in,
                              void* d_out, int out_size, void* d_ws, size_t ws_size,
                              hipStream_t stream) {
    const float* x   = (const float*)d_in[0];
    const float* h0  = (const float*)d_in[1];
    const float* c0  = (const float*)d_in[2];
    const float* w1  = (const float*)d_in[3];
    const float* b1  = (const float*)d_in[4];
    const float* wih = (const float*)d_in[5];
    const float* whh = (const float*)d_in[6];
    const float* bih = (const float*)d_in[7];
    const float* bhh = (const float*)d_in[8];
    const float* w2  = (const float*)d_in[9];
    const float* b2  = (const float*)d_in[10];
    float* out = (float*)d_out;

    dim3 grid(kB / 16), block(32);
    hipLaunchKernelGGL(policy_lstm_kernel, grid, block, 0, stream,
                       x, h0, c0, w1, b1, wih, whh, bih, bhh, w2, b2, out);
}